// SDFGuidedLoss_63324997812730
// MI455X (gfx1250) — compile-verified
//
#include <hip/hip_runtime.h>
#include <hip/hip_bf16.h>

// ---------------------------------------------------------------------------
// SDF-guided rotated-ROI-align loss, MI455X (gfx1250, wave32).
//
// Pipeline (all on `stream`, deterministic, no atomics):
//  K1: block-partial min/max of sdf_img (float4 streaming, 20.5MB -> ~1us HBM)
//  K2: final min/max -> affine (a,b) s.t. norm(v) = a*v + b   (folds the
//      [-1,1] normalization into the sampler: no 20MB materialization)
//  K3: one wave per ROI; lanes stride the 400 bilinear samples; per-wave
//      shfl reduction -> per-ROI masked sum  a*sum_raw + b*count_valid
//  K4: 16384 -> 1 reduction using V_WMMA_F32_16X16X4_F32 with all-ones A
//      (D[m][n] = sum_k B[k][n]; exact f32 sum of 64 values per WMMA)
// ---------------------------------------------------------------------------

typedef __attribute__((ext_vector_type(2))) float v2f;
typedef __attribute__((ext_vector_type(8))) float v8f;

#define N_ROI      16384
#define ROIS_PER_B 2048        // 16384 rois / 8 images
#define IMG_W      800
#define IMG_H      800
#define IMG_PIX    (IMG_W * IMG_H)
#define N_IMG_F4   1280000     // 8*800*800 / 4
#define MM_BLOCKS  512

// workspace layout (floats)
#define WS_AB      0           // [0]=a, [1]=b
#define WS_PMIN    64          // 512 block mins
#define WS_PMAX    576         // 512 block maxs
#define WS_ROISUM  4096        // 16384 per-ROI normalized sums

// ---------------- K1: partial min/max (streaming, float4) ------------------
__global__ void k_minmax_partial(const float* __restrict__ img,
                                 float* __restrict__ pmin,
                                 float* __restrict__ pmax) {
    int tid = blockIdx.x * blockDim.x + threadIdx.x;
    int stride = gridDim.x * blockDim.x;
    float mn = 3.402823466e38f, mx = -3.402823466e38f;
    const float4* p = (const float4*)img;
    for (int i = tid; i < N_IMG_F4; i += stride) {
        float4 v = p[i];
        mn = fminf(mn, fminf(fminf(v.x, v.y), fminf(v.z, v.w)));
        mx = fmaxf(mx, fmaxf(fmaxf(v.x, v.y), fmaxf(v.z, v.w)));
    }
    for (int off = 16; off; off >>= 1) {
        mn = fminf(mn, __shfl_down(mn, off));
        mx = fmaxf(mx, __shfl_down(mx, off));
    }
    __shared__ float smn[8], smx[8];
    int w = threadIdx.x >> 5, l = threadIdx.x & 31;
    if (l == 0) { smn[w] = mn; smx[w] = mx; }
    __syncthreads();
    if (w == 0) {
        mn = (l < 8) ? smn[l] : 3.402823466e38f;
        mx = (l < 8) ? smx[l] : -3.402823466e38f;
        for (int off = 4; off; off >>= 1) {
            mn = fminf(mn, __shfl_down(mn, off));
            mx = fmaxf(mx, __shfl_down(mx, off));
        }
        if (l == 0) { pmin[blockIdx.x] = mn; pmax[blockIdx.x] = mx; }
    }
}

// ---------------- K2: finish min/max, emit affine (a,b) --------------------
__global__ void k_minmax_final(const float* __restrict__ pmin,
                               const float* __restrict__ pmax,
                               float* __restrict__ ab) {
    int t = threadIdx.x;                       // 256 threads, 512 partials
    float mn = fminf(pmin[t], pmin[t + 256]);
    float mx = fmaxf(pmax[t], pmax[t + 256]);
    for (int off = 16; off; off >>= 1) {
        mn = fminf(mn, __shfl_down(mn, off));
        mx = fmaxf(mx, __shfl_down(mx, off));
    }
    __shared__ float smn[8], smx[8];
    int w = t >> 5, l = t & 31;
    if (l == 0) { smn[w] = mn; smx[w] = mx; }
    __syncthreads();
    if (t == 0) {
        float smin = smn[0], smax = smx[0];
        for (int i = 1; i < 8; ++i) {
            smin = fminf(smin, smn[i]);
            smax = fmaxf(smax, smx[i]);
        }
        float a, b;
        if (smax > smin) {
            float inv = 2.0f / (smax - smin);
            a = inv;
            b = -1.0f - smin * inv;
        } else { a = 1.0f; b = 0.0f; }
        ab[0] = a; ab[1] = b;
    }
}

// ---------------- K3: one wave per ROI, lanes stride 400 samples -----------
__global__ void __launch_bounds__(256) k_roi(const float* __restrict__ img,
                                             const float* __restrict__ bbox,
                                             const float* __restrict__ deg,
                                             const float* __restrict__ ab,
                                             float* __restrict__ roiSums) {
    const int wave = threadIdx.x >> 5;
    const int lane = threadIdx.x & 31;
    const int roi  = blockIdx.x * 8 + wave;        // 2048 blocks * 8 waves

    const float minx = bbox[roi * 4 + 0];
    const float miny = bbox[roi * 4 + 1];
    const float maxx = bbox[roi * 4 + 2];
    const float maxy = bbox[roi * 4 + 3];
    float cx  = 0.5f * (minx + maxx);
    float cy  = 0.5f * (miny + maxy);
    float wdt = maxx - minx;
    float hgt = maxy - miny;
    // jnp.clip(x, lo, hi) == min(max(x, lo), hi)  (hi wins when lo > hi)
    float wc  = fminf(fmaxf(wdt, 10.0f), 789.0f);
    float hc  = fminf(fmaxf(hgt, 10.0f), 789.0f);
    float cxc = fminf(fmaxf(cx, 0.5f * wc + 10.0f), 789.0f - 0.5f * wc);
    float cyc = fminf(fmaxf(cy, 0.5f * hc + 10.0f), 789.0f - 0.5f * hc);
    float theta = deg[roi] * 0.017453292519943295f;   // pi/180
    float st, ct;
    sincosf(theta, &st, &ct);
    cx = cxc - 0.5f;
    cy = cyc - 0.5f;

    const float* __restrict__ I = img + (roi >> 11) * IMG_PIX;  // roi/2048

    float sum = 0.0f, cnt = 0.0f;
    for (int j = lane; j < 400; j += 32) {          // 20x20 sample grid
        int iy = j / 20;
        int ix = j - iy * 20;
        float yy = ((float)iy + 0.5f) * 0.05f - 0.5f;
        float xx = ((float)ix + 0.5f) * 0.05f - 0.5f;
        yy *= hc;
        xx *= wc;
        float y = yy * ct - xx * st + cy;
        float x = yy * st + xx * ct + cx;
        bool valid = (y > -1.0f) && (y < 800.0f) && (x > -1.0f) && (x < 800.0f);
        y = fminf(fmaxf(y, 0.0f), 799.0f);
        x = fminf(fmaxf(x, 0.0f), 799.0f);
        float yf = floorf(y), xf = floorf(x);
        int y0 = (int)yf, x0 = (int)xf;
        int y1 = min(y0 + 1, 799), x1 = min(x0 + 1, 799);
        float ly = y - yf, lx = x - xf;
        float hy = 1.0f - ly, hx = 1.0f - lx;
        const float* r0 = I + y0 * IMG_W;
        const float* r1 = I + y1 * IMG_W;
        float v = hy * (hx * r0[x0] + lx * r0[x1]) +
                  ly * (hx * r1[x0] + lx * r1[x1]);
        if (valid) { sum += v; cnt += 1.0f; }
    }
    for (int off = 16; off; off >>= 1) {
        sum += __shfl_down(sum, off);
        cnt += __shfl_down(cnt, off);
    }
    if (lane == 0) {
        // sum of normalized, masked samples = a*sum_raw + b*count_valid
        roiSums[roi] = ab[0] * sum + ab[1] * cnt;
    }
}

// ---------------- K4: 16384 -> scalar via V_WMMA_F32_16X16X4_F32 -----------
// A = ones(16x4)  ->  D[m][n] = sum_k B[k][n] + C[m][n]; summing row 0's 16
// columns yields the exact f32 sum of all 64 B elements per WMMA, regardless
// of the lane->element mapping. 8 waves x 32 WMMA x 64 values = 16384.
__global__ void __launch_bounds__(256) k_final(const float* __restrict__ roiSums,
                                               float* __restrict__ out) {
    const int lane = threadIdx.x & 31;
    const int w    = threadIdx.x >> 5;             // 0..7
    v2f a; a.x = 1.0f; a.y = 1.0f;                 // all-ones A (16x4 = 2 VGPRs)
    v8f c = {};
    const float* __restrict__ base = roiSums + w * 2048;
    for (int i = 0; i < 32; ++i) {
        v2f b;
        b.x = base[i * 64 + lane];
        b.y = base[i * 64 + 32 + lane];
        c = __builtin_amdgcn_wmma_f32_16x16x4_f32(
                /*neg_a=*/false, a, /*neg_b=*/false, b,
                /*c_mod=*/(short)0, c, /*reuse_a=*/false, /*reuse_b=*/false);
    }
    // c[0]: lanes 0..15 hold (M=0, N=lane); lanes 16..31 hold (M=8, N=lane-16).
    // All D rows are identical column sums -> sum lanes 0..15 only.
    float val = (lane < 16) ? c[0] : 0.0f;
    for (int off = 16; off; off >>= 1) val += __shfl_down(val, off);
    __shared__ float sacc[8];
    if (lane == 0) sacc[w] = val;
    __syncthreads();
    if (threadIdx.x == 0) {
        float t = 0.0f;
        for (int i = 0; i < 8; ++i) t += sacc[i];
        float mean = t / 6553600.0f;               // 16384 rois * 400 samples
        out[0] = (mean + 100.0f) / 200.0f;         // (loss + rng) / (2*rng)
    }
}

// ---------------------------------------------------------------------------
extern "C" void kernel_launch(void* const* d_in, const int* in_sizes, int n_in,
                              void* d_out, int out_size, void* d_ws, size_t ws_size,
                              hipStream_t stream) {
    const float* sdf_img = (const float*)d_in[0];   // (8,1,800,800) f32
    const float* bbox    = (const float*)d_in[1];   // (16384,4) f32
    const float* deg     = (const float*)d_in[2];   // (16384,)  f32
    float* out = (float*)d_out;
    float* wsF = (float*)d_ws;

    float* ab      = wsF + WS_AB;
    float* pmin    = wsF + WS_PMIN;
    float* pmax    = wsF + WS_PMAX;
    float* roiSums = wsF + WS_ROISUM;

    k_minmax_partial<<<MM_BLOCKS, 256, 0, stream>>>(sdf_img, pmin, pmax);
    k_minmax_final<<<1, 256, 0, stream>>>(pmin, pmax, ab);
    k_roi<<<N_ROI / 8, 256, 0, stream>>>(sdf_img, bbox, deg, ab, roiSums);
    k_final<<<1, 256, 0, stream>>>(roiSums, out);
}